// GConv_53455162966381
// MI455X (gfx1250) — compile-verified
//
#include <hip/hip_runtime.h>
#include <hip/hip_bf16.h>

typedef __attribute__((ext_vector_type(2))) float v2f;
typedef __attribute__((ext_vector_type(8))) float v8f;

#define HIDF 128   // output feature width of every GEMM in this net

// ---------------------------------------------------------------------------
// Degree / normalization kernels
// ---------------------------------------------------------------------------
__global__ void k_init_deg(float* __restrict__ deg, int N) {
    int i = blockIdx.x * blockDim.x + threadIdx.x;
    if (i < N) deg[i] = 1.0f;               // self-loop contributes 1
}

__global__ void k_count_deg(float* __restrict__ deg, const int* __restrict__ cols, int E) {
    int e = blockIdx.x * blockDim.x + threadIdx.x;
    if (e < E) atomicAdd(&deg[cols[e]], 1.0f);
}

__global__ void k_rsqrt_deg(float* __restrict__ deg, int N) {
    int i = blockIdx.x * blockDim.x + threadIdx.x;
    if (i < N) deg[i] = rsqrtf(deg[i]);     // deg >= 1 always (self loops)
}

// ---------------------------------------------------------------------------
// GEMM: C[M x 128] = A[M x K] * B[K x 128] (+ bias), fp32 WMMA 16x16x4
// One wave per 16x16 tile; 8 waves per block cover the full 128-wide row.
// ---------------------------------------------------------------------------
__global__ void __launch_bounds__(256)
k_gemm_wmma_f32(const float* __restrict__ A, const float* __restrict__ B,
                float* __restrict__ C, const float* __restrict__ bias,
                int M, int K) {
    const int lane = threadIdx.x & 31;
    const int wave = threadIdx.x >> 5;
    const int m0   = blockIdx.x * 16;
    const int n0   = wave * 16;
    const int hi   = lane >> 4;     // 0: lanes 0-15, 1: lanes 16-31
    const int r    = lane & 15;

    v8f acc = {0.f, 0.f, 0.f, 0.f, 0.f, 0.f, 0.f, 0.f};

    // A row for this lane (clamp keeps loads in-bounds without divergence)
    const int arow = (m0 + r < M) ? (m0 + r) : (M - 1);
    const float* __restrict__ Ar = A + (size_t)arow * K;

    #pragma unroll 4
    for (int k = 0; k < K; k += 4) {
        v2f a, b;
        // A 16x4: lanes 0-15 hold K=k..k+1, lanes 16-31 hold K=k+2..k+3
        a.x = Ar[k + 2 * hi];
        a.y = Ar[k + 2 * hi + 1];
        // B 4x16: VGPR0 -> rows {k, k+2}, VGPR1 -> rows {k+1, k+3}, col = n0+r
        b.x = B[(size_t)(k + 2 * hi)     * HIDF + n0 + r];
        b.y = B[(size_t)(k + 2 * hi + 1) * HIDF + n0 + r];
        acc = __builtin_amdgcn_wmma_f32_16x16x4_f32(
                  false, a, false, b, (short)0, acc, false, false);
    }

    const float badd = bias ? bias[n0 + r] : 0.0f;
    #pragma unroll
    for (int i = 0; i < 8; ++i) {
        int m = m0 + i + 8 * hi;           // C/D: VGPR i -> M = i + 8*hi
        if (m < M) C[(size_t)m * HIDF + n0 + r] = acc[i] + badd;
    }
}

// ---------------------------------------------------------------------------
// Z[i] = bias + dinv[i]^2 * H[i]   (self-loop message + bias), float4 lanes
// ---------------------------------------------------------------------------
__global__ void k_selfloop_bias(const float4* __restrict__ H4,
                                const float* __restrict__ dinv,
                                const float* __restrict__ bias,
                                float4* __restrict__ Z4, int N) {
    int t    = blockIdx.x * blockDim.x + threadIdx.x;
    int node = t >> 5;                     // 32 float4 = 128 floats per node
    int lane = t & 31;
    if (node >= N) return;
    float d = dinv[node];
    float s = d * d;
    float4 v = H4[(size_t)node * 32 + lane];
    float4 b = reinterpret_cast<const float4*>(bias)[lane];
    float4 o;
    o.x = b.x + s * v.x;
    o.y = b.y + s * v.y;
    o.z = b.z + s * v.z;
    o.w = b.w + s * v.w;
    Z4[(size_t)node * 32 + lane] = o;
}

// ---------------------------------------------------------------------------
// Edge scatter: one wave per edge; 512B coalesced gather from L2-resident H,
// 4 global f32 atomic adds per lane into the target row.
// ---------------------------------------------------------------------------
__global__ void __launch_bounds__(256)
k_edge_agg(const float4* __restrict__ H4, float* __restrict__ Out,
           const int* __restrict__ rows, const int* __restrict__ cols,
           const float* __restrict__ dinv, int E) {
    int widx = (int)((blockIdx.x * blockDim.x + threadIdx.x) >> 5);
    int lane = threadIdx.x & 31;
    if (widx >= E) return;                 // uniform per wave
    int src = rows[widx];
    int dst = cols[widx];
    float norm = dinv[src] * dinv[dst];
    float4 v = H4[(size_t)src * 32 + lane];
    float* o = Out + (size_t)dst * HIDF + lane * 4;
    atomicAdd(o + 0, v.x * norm);
    atomicAdd(o + 1, v.y * norm);
    atomicAdd(o + 2, v.z * norm);
    atomicAdd(o + 3, v.w * norm);
}

// ---------------------------------------------------------------------------
extern "C" void kernel_launch(void* const* d_in, const int* in_sizes, int n_in,
                              void* d_out, int out_size, void* d_ws, size_t ws_size,
                              hipStream_t stream) {
    const float* x  = (const float*)d_in[0];   // [N, 256]
    const int*   ei = (const int*)  d_in[1];   // [2, E]
    const float* W1 = (const float*)d_in[2];   // [256, 128]
    const float* b1 = (const float*)d_in[3];   // [128]
    const float* W2 = (const float*)d_in[4];   // [128, 128]
    const float* b2 = (const float*)d_in[5];   // [128]
    const float* Wp = (const float*)d_in[6];   // [128, 128]
    const float* bp = (const float*)d_in[7];   // [128]

    const int N = in_sizes[0] / 256;
    const int E = in_sizes[1] / 2;
    const int* rows = ei;                      // edge_index[0] = sources
    const int* cols = ei + E;                  // edge_index[1] = targets

    // Workspace: [dinv (N, padded to 64) | H (N x 128)]
    float* dinv = (float*)d_ws;
    float* H    = dinv + (((size_t)N + 63) & ~(size_t)63);

    // d_out: [ z (N x 128) | proj (N x 128) ]; second half doubles as Z1 scratch
    float* z_out = (float*)d_out;
    float* proj  = z_out + (size_t)N * HIDF;

    const int TB = 256;
    dim3 blkN((N + TB - 1) / TB);
    dim3 blkE((E + TB - 1) / TB);
    dim3 blkNode(((size_t)N * 32 + TB - 1) / TB);  // N*32 float4 threads
    dim3 blkEdge((E + 7) / 8);                     // 8 waves (edges) per block
    dim3 blkGemm((N + 15) / 16);

    // 1) normalization coefficients
    k_init_deg <<<blkN, TB, 0, stream>>>(dinv, N);
    k_count_deg<<<blkE, TB, 0, stream>>>(dinv, cols, E);
    k_rsqrt_deg<<<blkN, TB, 0, stream>>>(dinv, N);

    // 2) layer 1: H = x @ W1 ; Z1 = aggregate(H) + b1   (Z1 lives in proj region)
    k_gemm_wmma_f32<<<blkGemm, TB, 0, stream>>>(x, W1, H, nullptr, N, 256);
    k_selfloop_bias<<<blkNode, TB, 0, stream>>>((const float4*)H, dinv, b1,
                                                (float4*)proj, N);
    k_edge_agg<<<blkEdge, TB, 0, stream>>>((const float4*)H, proj, rows, cols, dinv, E);

    // 3) layer 2: H = Z1 @ W2 ; z = aggregate(H) + b2
    k_gemm_wmma_f32<<<blkGemm, TB, 0, stream>>>(proj, W2, H, nullptr, N, HIDF);
    k_selfloop_bias<<<blkNode, TB, 0, stream>>>((const float4*)H, dinv, b2,
                                                (float4*)z_out, N);
    k_edge_agg<<<blkEdge, TB, 0, stream>>>((const float4*)H, z_out, rows, cols, dinv, E);

    // 4) projection head: proj = z @ Wp + bp  (overwrites Z1 scratch)
    k_gemm_wmma_f32<<<blkGemm, TB, 0, stream>>>(z_out, Wp, proj, bp, N, HIDF);
}